// Model_8830452760682
// MI455X (gfx1250) — compile-verified
//
#include <hip/hip_runtime.h>
#include <math.h>
#include <stdint.h>

#define BSZ 128
#define SEQ 64
#define DIM 128
#define NTOT (BSZ * (SEQ + 1))   // 8320
#define ROWS (BSZ * SEQ)         // 8192
#define NEGV (-10000.0f)
#define ALPHA_C (5e-05f)

typedef __attribute__((ext_vector_type(2))) float v2f;
typedef __attribute__((ext_vector_type(8))) float v8f;

// issue one async 16B/lane memory->LDS copy; IOFFSET applies to both sides
__device__ __forceinline__ void async_b128(uint32_t lds_off, const float* gaddr, int imm) {
    switch (imm) {
    case 0:
        asm volatile("global_load_async_to_lds_b128 %0, %1, off offset:0"
                     :: "v"(lds_off), "v"(gaddr) : "memory");
        break;
    case 4096:
        asm volatile("global_load_async_to_lds_b128 %0, %1, off offset:4096"
                     :: "v"(lds_off), "v"(gaddr) : "memory");
        break;
    case 8192:
        asm volatile("global_load_async_to_lds_b128 %0, %1, off offset:8192"
                     :: "v"(lds_off), "v"(gaddr) : "memory");
        break;
    default:
        asm volatile("global_load_async_to_lds_b128 %0, %1, off offset:12288"
                     :: "v"(lds_off), "v"(gaddr) : "memory");
        break;
    }
}
__device__ __forceinline__ void wait_async0() {
    asm volatile("s_wait_asynccnt 0x0" ::: "memory");
}

// ---------------------------------------------------------------- gather
__global__ void gather_kernel(const int* __restrict__ items,
                              const float* __restrict__ id_table,
                              const float* __restrict__ pop,
                              float* __restrict__ score,
                              float* __restrict__ debias,
                              unsigned char* __restrict__ zf) {
    int r = blockIdx.x;
    int it = items[r];
    score[r * DIM + threadIdx.x] = id_table[(long)it * DIM + threadIdx.x];
    if (threadIdx.x == 0) {
        debias[r] = logf(pop[it]);
        zf[r] = (it == 0) ? 1 : 0;
    }
}

// ------------------------------------------------------- sequence membership
__global__ __launch_bounds__(256) void inseq_kernel(const int* __restrict__ items,
                                                    unsigned char* __restrict__ inseq) {
    __shared__ int seq[SEQ + 1];
    int b = blockIdx.x;
    if (threadIdx.x < SEQ + 1) seq[threadIdx.x] = items[b * (SEQ + 1) + threadIdx.x];
    __syncthreads();
    for (int j = threadIdx.x; j < NTOT; j += blockDim.x) {
        int it = items[j];
        bool f = false;
        for (int s = 0; s < SEQ + 1; ++s) f = f || (seq[s] == it);
        inseq[b * NTOT + j] = f ? 1 : 0;
    }
}

// ------------------------------------------------- prec = embs[:, :-1] @ W_u
__global__ __launch_bounds__(256) void prec_gemm(const float* __restrict__ score,
                                                 const float* __restrict__ Wu,
                                                 float* __restrict__ prec) {
    __shared__ float As[16 * DIM];
    const int tid = threadIdx.x;
    const int wave = tid >> 5;      // column tile 0..7
    const int lane = tid & 31;
    const int half = lane >> 4;
    const int ln = lane & 15;
    const int rowBase = blockIdx.x * 16;

    // vectorized A staging (remap rows to skip last seq position per batch)
    float4* Asv = (float4*)As;
    for (int idx = tid; idx < 16 * (DIM / 4); idx += 256) {
        int rl = idx >> 5;                 // row 0..15
        int d4 = idx & 31;                 // float4 within row
        int r = rowBase + rl;
        int src = (r >> 6) * 65 + (r & 63);
        Asv[idx] = ((const float4*)score)[src * (DIM / 4) + d4];
    }
    __syncthreads();

    v8f c = {};
    const float* Ap = &As[ln * DIM + 2 * half];
#pragma unroll
    for (int k = 0; k < 32; ++k) {
        v2f av, bv;
        av.x = Ap[4 * k];
        av.y = Ap[4 * k + 1];
        int k0 = 4 * k + 2 * half;
        bv.x = Wu[k0 * DIM + wave * 16 + ln];
        bv.y = Wu[(k0 + 1) * DIM + wave * 16 + ln];
        c = __builtin_amdgcn_wmma_f32_16x16x4_f32(false, av, false, bv,
                                                  (short)0, c, false, false);
    }
#pragma unroll
    for (int i = 0; i < 8; ++i) {
        int row = rowBase + i + 8 * half;
        int col = wave * 16 + ln;
        prec[row * DIM + col] = c[i];
    }
}

// ------------- fused logits GEMM + mask + online log-softmax + CE reduction
// A fragments register-resident; B panels double-buffered via async-to-LDS.
__global__ __launch_bounds__(256) void ce_kernel(const float* __restrict__ prec,
                                                 const float* __restrict__ score,
                                                 const float* __restrict__ debias,
                                                 const unsigned char* __restrict__ inseq,
                                                 const unsigned char* __restrict__ zf,
                                                 const int* __restrict__ logmask,
                                                 float* __restrict__ acc) {
    __shared__ float smem[2 * 32 * DIM];    // 32 KB: A staging, then B double-buffer
    __shared__ float m_sh[2][64], l_sh[2][64], lab_sh[2][64];

    const int tid = threadIdx.x;
    const int wave = tid >> 5;
    const int lane = tid & 31;
    const int half = lane >> 4;
    const int ln = lane & 15;
    const int rt = wave >> 1;               // row tile 0..3
    const int ct = wave & 1;                // col tile 0..1
    const int bb = blockIdx.x;              // one batch per block
    const int rowBase = bb * 64;

    // ---- stage A panel (64x128) once, vectorized, then lift to registers
    {
        const float4* precv = (const float4*)(prec + (size_t)rowBase * DIM);
        float4* smemv = (float4*)smem;
        for (int idx = tid; idx < 64 * (DIM / 4); idx += 256)
            smemv[idx] = precv[idx];
    }
    __syncthreads();
    v2f a_reg[32];
    {
        const float* Ap = &smem[(rt * 16 + ln) * DIM + 2 * half];
#pragma unroll
        for (int k = 0; k < 32; ++k) {
            a_reg[k].x = Ap[4 * k];
            a_reg[k].y = Ap[4 * k + 1];
        }
    }
    __syncthreads();

    float* Bs0 = smem;
    float* Bs1 = smem + 32 * DIM;
    const float* Bp0 = &Bs0[(ct * 16 + ln) * DIM + 2 * half];
    const float* Bp1 = &Bs1[(ct * 16 + ln) * DIM + 2 * half];

    // ---- prologue: async-stage panel 0
    {
        uint32_t l0 = (uint32_t)(uintptr_t)(Bs0 + tid * 4);
        const float* g0 = score + tid * 4;
        async_b128(l0, g0, 0);
        async_b128(l0, g0, 4096);
        async_b128(l0, g0, 8192);
        async_b128(l0, g0, 12288);
    }
    wait_async0();
    __syncthreads();

    float m_run[8], l_run[8], labv[8];
#pragma unroll
    for (int i = 0; i < 8; ++i) { m_run[i] = -INFINITY; l_run[i] = 0.f; labv[i] = -INFINITY; }

    for (int j0 = 0; j0 < NTOT; j0 += 32) {
        const int buf = (j0 >> 5) & 1;

        // issue async load of next panel into the other buffer
        if (j0 + 32 < NTOT) {
            float* nb = buf ? Bs0 : Bs1;
            uint32_t l0 = (uint32_t)(uintptr_t)(nb + tid * 4);
            const float* g0 = score + (size_t)(j0 + 32) * DIM + tid * 4;
            async_b128(l0, g0, 0);
            async_b128(l0, g0, 4096);
            async_b128(l0, g0, 8192);
            async_b128(l0, g0, 12288);
        }

        const float* Bp = buf ? Bp1 : Bp0;
        v8f c = {};
#pragma unroll
        for (int k = 0; k < 32; ++k) {
            v2f bv;
            bv.x = Bp[4 * k];
            bv.y = Bp[4 * k + 1];
            c = __builtin_amdgcn_wmma_f32_16x16x4_f32(false, a_reg[k], false, bv,
                                                      (short)0, c, false, false);
        }

        const int j = j0 + ct * 16 + ln;              // global column of this lane
        const float dbj = debias[j];
        const bool hist = (zf[j] != 0) || (inseq[bb * NTOT + j] != 0);

#pragma unroll
        for (int i = 0; i < 8; ++i) {
            const int s = rt * 16 + i + 8 * half;     // 0..63
            const int label = bb * 65 + s + 1;
            float val = c[i] - dbj;
            if (hist && (j != label)) val = NEGV;
            if (j == label) labv[i] = val;

            float vmax = val;
            for (int off = 1; off < 16; off <<= 1)
                vmax = fmaxf(vmax, __shfl_xor(vmax, off, 32));
            float mnew = fmaxf(m_run[i], vmax);
            float e = expf(val - mnew);
            for (int off = 1; off < 16; off <<= 1)
                e += __shfl_xor(e, off, 32);
            l_run[i] = l_run[i] * expf(m_run[i] - mnew) + e;
            m_run[i] = mnew;
        }

        wait_async0();
        __syncthreads();
    }

#pragma unroll
    for (int i = 0; i < 8; ++i) {
        float lm = labv[i];
        for (int off = 1; off < 16; off <<= 1)
            lm = fmaxf(lm, __shfl_xor(lm, off, 32));
        if (ln == 0) {
            int rl = rt * 16 + i + 8 * half;
            m_sh[ct][rl] = m_run[i];
            l_sh[ct][rl] = l_run[i];
            lab_sh[ct][rl] = lm;
        }
    }
    __syncthreads();
    if (tid < 64) {
        int row = rowBase + tid;
        float m0 = m_sh[0][tid], m1 = m_sh[1][tid];
        float mm = fmaxf(m0, m1);
        float l = l_sh[0][tid] * expf(m0 - mm) + l_sh[1][tid] * expf(m1 - mm);
        float lab = fmaxf(lab_sh[0][tid], lab_sh[1][tid]);
        float ce = (mm + logf(l)) - lab;
        float w = (logmask[row] != 0) ? 1.f : 0.f;
        atomicAdd(&acc[0], ce * w);
        atomicAdd(&acc[1], w);
    }
}

// ------------------------------------------------------------ batchnorm stats
__global__ void bn_stats(const float* __restrict__ x, int rows,
                         float* __restrict__ sum, float* __restrict__ sumsq) {
    int d = threadIdx.x;
    float a = 0.f, a2 = 0.f;
    for (int r = blockIdx.x; r < rows; r += gridDim.x) {
        float v = x[r * DIM + d];
        a += v;
        a2 += v * v;
    }
    atomicAdd(&sum[d], a);
    atomicAdd(&sumsq[d], a2);
}

__global__ void bn_finalize(const float* __restrict__ sum, const float* __restrict__ sumsq,
                            float rows, float* __restrict__ mu, float* __restrict__ rstd) {
    int d = threadIdx.x;
    float m = sum[d] / rows;
    float v = sumsq[d] / rows - m * m;
    mu[d] = m;
    rstd[d] = rsqrtf(v + 1e-5f);
}

// ------------------------------------------- batchnorm + row L2-normalization
__global__ void bnnorm_kernel(const float* __restrict__ x, const float* __restrict__ mu,
                              const float* __restrict__ rstd, const float* __restrict__ gamma,
                              const float* __restrict__ beta, float* __restrict__ out) {
    __shared__ float red[DIM];
    int r = blockIdx.x, d = threadIdx.x;
    float v = (x[r * DIM + d] - mu[d]) * rstd[d] * gamma[d] + beta[d];
    red[d] = v * v;
    __syncthreads();
    for (int off = 64; off > 0; off >>= 1) {
        if (d < off) red[d] += red[d + off];
        __syncthreads();
    }
    float n = sqrtf(red[0]);
    out[r * DIM + d] = v / fmaxf(n, 1e-12f);
}

// ----------------------------------------------- cosine-sim argmax per batch
__global__ __launch_bounds__(256) void cosmax_kernel(const float* __restrict__ flatin,
                                                     const unsigned char* __restrict__ inseq,
                                                     int* __restrict__ closest) {
    __shared__ float tgt[DIM];
    __shared__ float sval[256];
    __shared__ int sidx[256];
    int b = blockIdx.x;
    int tid = threadIdx.x;
    for (int d = tid; d < DIM; d += 256) tgt[d] = flatin[(b * 65 + 64) * DIM + d];
    __syncthreads();
    float al2 = 0.f;
    for (int d = 0; d < DIM; ++d) al2 += tgt[d] * tgt[d];
    float best = -INFINITY;
    int bestj = 0;
    for (int j = tid; j < NTOT; j += 256) {
        if (inseq[b * NTOT + j]) continue;
        float num = 0.f, bl2 = 0.f;
        for (int d = 0; d < DIM; ++d) {
            float v = flatin[j * DIM + d];
            num += tgt[d] * v;
            bl2 += v * v;
        }
        float den = fmaxf(sqrtf(al2 * bl2), 1e-8f);
        float s = num / den;
        if (s > best) { best = s; bestj = j; }
    }
    sval[tid] = best;
    sidx[tid] = bestj;
    __syncthreads();
    for (int off = 128; off > 0; off >>= 1) {
        if (tid < off) {
            if (sval[tid + off] > sval[tid] ||
                (sval[tid + off] == sval[tid] && sidx[tid + off] < sidx[tid])) {
                sval[tid] = sval[tid + off];
                sidx[tid] = sidx[tid + off];
            }
        }
        __syncthreads();
    }
    if (tid == 0) closest[b] = sidx[0];
}

// --------------------------------------------------------------------- mi
__global__ void mi_kernel(const float* __restrict__ histe, const float* __restrict__ flatin,
                          const int* __restrict__ closest, float* __restrict__ acc) {
    __shared__ float red[DIM];
    int row = blockIdx.x, d = threadIdx.x;
    int b = row >> 6;
    float diff = histe[row * DIM + d] - flatin[closest[b] * DIM + d];
    red[d] = diff * diff;
    __syncthreads();
    for (int off = 64; off > 0; off >>= 1) {
        if (d < off) red[d] += red[d + off];
        __syncthreads();
    }
    if (d == 0) atomicAdd(&acc[2], sqrtf(red[0]) * (1.0f / (float)SEQ));
}

// --------------------------------------------------- alignment + un/itn prep
__global__ void align_kernel(const float* __restrict__ prec, const float* __restrict__ score,
                             float* __restrict__ un, float* __restrict__ itn,
                             float* __restrict__ acc) {
    __shared__ float red[DIM];
    int b = blockIdx.x, d = threadIdx.x;
    float u = prec[(b * 64 + 63) * DIM + d];
    float it = score[(b * 65 + 64) * DIM + d];

    red[d] = u * u;
    __syncthreads();
    for (int off = 64; off > 0; off >>= 1) { if (d < off) red[d] += red[d + off]; __syncthreads(); }
    float nu = fmaxf(sqrtf(red[0]), 1e-12f);
    __syncthreads();

    red[d] = it * it;
    __syncthreads();
    for (int off = 64; off > 0; off >>= 1) { if (d < off) red[d] += red[d + off]; __syncthreads(); }
    float ni = fmaxf(sqrtf(red[0]), 1e-12f);
    __syncthreads();

    float unv = u / nu, itv = it / ni;
    un[b * DIM + d] = unv;
    itn[b * DIM + d] = itv;
    float df = unv - itv;
    red[d] = df * df;
    __syncthreads();
    for (int off = 64; off > 0; off >>= 1) { if (d < off) red[d] += red[d + off]; __syncthreads(); }
    if (d == 0) atomicAdd(&acc[3], red[0] * (1.0f / (float)BSZ));
}

// ------------------------------------------------------------- uniformity
__global__ void uniform_kernel(const float* __restrict__ un, const float* __restrict__ itn,
                               float* __restrict__ acc) {
    __shared__ float xi[DIM];
    __shared__ float red[DIM];
    int i = blockIdx.x;
    int which = blockIdx.y;
    const float* x = which ? itn : un;
    int tid = threadIdx.x;
    xi[tid] = x[i * DIM + tid];
    __syncthreads();
    float e = 0.f;
    int j = tid;
    if (j > i) {
        float sq = 0.f;
        for (int d = 0; d < DIM; ++d) {
            float df = xi[d] - x[j * DIM + d];
            sq += df * df;
        }
        e = expf(-2.0f * sq);
    }
    red[tid] = e;
    __syncthreads();
    for (int off = 64; off > 0; off >>= 1) { if (tid < off) red[tid] += red[tid + off]; __syncthreads(); }
    if (tid == 0) atomicAdd(&acc[4 + which], red[0]);
}

// ------------------------------------------------------------- init/finalize
__global__ void init_kernel(float* __restrict__ acc, float* __restrict__ bnsum) {
    int t = threadIdx.x;
    if (t < 8) acc[t] = 0.f;
    if (t < 512) bnsum[t] = 0.f;
}

__global__ void finalize_kernel(const float* __restrict__ acc, float* __restrict__ out) {
    if (threadIdx.x == 0) {
        float loss = acc[0] / acc[1];
        float npairs = (float)(BSZ * (BSZ - 1) / 2);   // 8128
        out[0] = loss + ALPHA_C * acc[2];
        out[1] = acc[3];
        out[2] = 0.5f * (logf(acc[4] / npairs) + logf(acc[5] / npairs));
    }
}

// ----------------------------------------------------------------------------
extern "C" void kernel_launch(void* const* d_in, const int* in_sizes, int n_in,
                              void* d_out, int out_size, void* d_ws, size_t ws_size,
                              hipStream_t stream) {
    (void)in_sizes; (void)n_in; (void)out_size; (void)ws_size;
    const int*   items   = (const int*)d_in[0];
    const int*   logmask = (const int*)d_in[1];
    const float* idtab   = (const float*)d_in[2];
    const float* pop     = (const float*)d_in[3];
    const float* Wu      = (const float*)d_in[4];
    const float* gamma   = (const float*)d_in[5];
    const float* beta    = (const float*)d_in[6];
    float* out = (float*)d_out;

    // workspace layout (floats, then bytes)
    float* ws     = (float*)d_ws;
    float* score  = ws;                           // NTOT*DIM
    float* prec   = score + (size_t)NTOT * DIM;   // ROWS*DIM
    float* flatin = prec + (size_t)ROWS * DIM;    // NTOT*DIM
    float* histe  = flatin + (size_t)NTOT * DIM;  // ROWS*DIM
    float* debias = histe + (size_t)ROWS * DIM;   // NTOT
    float* un     = debias + NTOT;                // BSZ*DIM
    float* itn    = un + BSZ * DIM;               // BSZ*DIM
    float* bnsum  = itn + BSZ * DIM;              // 8*128
    float* sum1 = bnsum, *sq1 = bnsum + 128, *sum2 = bnsum + 256, *sq2 = bnsum + 384;
    float* mu1 = bnsum + 512, *rstd1 = bnsum + 640, *mu2 = bnsum + 768, *rstd2 = bnsum + 896;
    float* acc = bnsum + 1024;                    // 8 accumulators
    int* closest = (int*)(acc + 8);               // BSZ ints
    unsigned char* zf    = (unsigned char*)(closest + BSZ);   // NTOT bytes
    unsigned char* inseq = zf + NTOT;                         // BSZ*NTOT bytes

    init_kernel<<<1, 512, 0, stream>>>(acc, bnsum);
    gather_kernel<<<NTOT, DIM, 0, stream>>>(items, idtab, pop, score, debias, zf);
    inseq_kernel<<<BSZ, 256, 0, stream>>>(items, inseq);
    prec_gemm<<<ROWS / 16, 256, 0, stream>>>(score, Wu, prec);
    ce_kernel<<<BSZ, 256, 0, stream>>>(prec, score, debias, inseq, zf, logmask, acc);

    bn_stats<<<64, DIM, 0, stream>>>(score, NTOT, sum1, sq1);
    bn_stats<<<64, DIM, 0, stream>>>(prec, ROWS, sum2, sq2);
    bn_finalize<<<1, DIM, 0, stream>>>(sum1, sq1, (float)NTOT, mu1, rstd1);
    bn_finalize<<<1, DIM, 0, stream>>>(sum2, sq2, (float)ROWS, mu2, rstd2);
    bnnorm_kernel<<<NTOT, DIM, 0, stream>>>(score, mu1, rstd1, gamma, beta, flatin);
    bnnorm_kernel<<<ROWS, DIM, 0, stream>>>(prec, mu2, rstd2, gamma, beta, histe);

    align_kernel<<<BSZ, DIM, 0, stream>>>(prec, score, un, itn, acc);
    cosmax_kernel<<<BSZ, 256, 0, stream>>>(flatin, inseq, closest);
    mi_kernel<<<ROWS, DIM, 0, stream>>>(histe, flatin, closest, acc);
    uniform_kernel<<<dim3(BSZ, 2), DIM, 0, stream>>>(un, itn, acc);
    finalize_kernel<<<1, 32, 0, stream>>>(acc, out);
}